// seqLSTM_1889785610251
// MI455X (gfx1250) — compile-verified
//
#include <hip/hip_runtime.h>
#include <math.h>

typedef __attribute__((ext_vector_type(2))) float v2f;
typedef __attribute__((ext_vector_type(8))) float v8f;
typedef __attribute__((ext_vector_type(4))) unsigned int v4u;
typedef __attribute__((ext_vector_type(8))) int v8i;
typedef __attribute__((ext_vector_type(4))) int v4i;

// ---- workspace layout (float offsets) ----
#define WS_XW      0u          // 2 * 2048 * 1024
#define WS_OUTBUF  4194304u    // 32*64*512
#define WS_PART    5242880u    // 64*16*512
#define WS_OUTF    5767168u    // 64*512
#define WS_LEN     5799936u    // 32 ints

// ---------- WMMA f32 16x16x4 fragment helpers ----------
// A (16x4, MxK): lanes 0-15 hold rows m0+l, K={k,k+1}; lanes 16-31 rows m0+l-16, K={k+2,k+3}
__device__ __forceinline__ v2f load_a4(const float* A, int lda, int m0, int k, int lane) {
  int half = lane >> 4, r = lane & 15;
  const float* p = A + (size_t)(m0 + r) * lda + k + 2 * half;
  v2f a; a.x = p[0]; a.y = p[1]; return a;
}
// B (4x16, KxN): lanes 0-15 hold col n0+l, K={k,k+1}; lanes 16-31 col n0+l-16, K={k+2,k+3}
__device__ __forceinline__ v2f load_b4(const float* B, int ldb, int k, int n0, int lane) {
  int half = lane >> 4, c = lane & 15;
  const float* p = B + (size_t)(k + 2 * half) * ldb + n0 + c;
  v2f b; b.x = p[0]; b.y = p[ldb]; return b;
}
__device__ __forceinline__ v8f wmma4(v2f a, v2f b, v8f c) {
  return __builtin_amdgcn_wmma_f32_16x16x4_f32(false, a, false, b, (short)0, c, false, false);
}
__device__ __forceinline__ float sigf(float x) { return 1.0f / (1.0f + expf(-x)); }

// ---------- Tensor Data Mover: 2-D tile (row-major, f32) global -> LDS ----------
// Descriptor per CDNA5 ISA §8.3/§8.4: group0 = {count/type, lds_addr, global_addr},
// group1 = {data_size, tensor_dim0/1, tile_dim0/1, tensor_dim0_stride}; groups 2/3 zero (2-D).
__device__ __forceinline__ void tdm_load_2d(unsigned lds_off, const float* gaddr,
                                            unsigned tensor_d0, unsigned tensor_d1,
                                            unsigned tile_d0, unsigned tile_d1,
                                            unsigned stride0) {
  unsigned long long ga = (unsigned long long)(uintptr_t)gaddr;
  v4u g0;
  g0[0] = 1u;                                               // count=1, user desc
  g0[1] = lds_off;                                          // LDS byte address
  g0[2] = (unsigned)(ga & 0xFFFFFFFFu);                     // global_addr[31:0]
  g0[3] = (unsigned)((ga >> 32) & 0x1FFFFFFu) | (2u << 30); // global_addr[56:32], type=2
  v8i g1;
  g1[0] = (int)(2u << 16);                                  // data_size = 4 bytes
  g1[1] = (int)((tensor_d0 & 0xFFFFu) << 16);               // tensor_dim0[15:0]
  g1[2] = (int)((tensor_d0 >> 16) | ((tensor_d1 & 0xFFFFu) << 16));
  g1[3] = (int)((tensor_d1 >> 16) | (tile_d0 << 16));       // tile_dim0
  g1[4] = (int)(tile_d1 & 0xFFFFu);                         // tile_dim1 (tile_dim2=0)
  g1[5] = (int)stride0;                                     // tensor_dim0_stride[31:0]
  g1[6] = 0;
  g1[7] = 0;
  v4i z4 = {0, 0, 0, 0};
#if __clang_major__ >= 23
  v8i z8 = {0, 0, 0, 0, 0, 0, 0, 0};
  __builtin_amdgcn_tensor_load_to_lds(g0, g1, z4, z4, z8, 0);
#else
  __builtin_amdgcn_tensor_load_to_lds(g0, g1, z4, z4, 0);
#endif
}
__device__ __forceinline__ unsigned lds_off_of(const void* p) {
  return (unsigned)(uintptr_t)p;   // LDS flat address: offset lives in the low 32 bits
}

// ---------- Kernel A: sequence lengths ----------
__global__ void k_len(const float* __restrict__ x, int* __restrict__ len) {
  int b = blockIdx.x, t = threadIdx.x; // 32 blocks x 64 threads
  const float* p = x + ((size_t)b * 64 + t) * 300;
  int used = 0;
  for (int d = 0; d < 300; ++d) if (p[d] != 0.0f) { used = 1; break; }
  __shared__ int s[64];
  s[t] = used;
  __syncthreads();
  if (t == 0) { int sum = 0; for (int i = 0; i < 64; ++i) sum += s[i]; len[b] = sum; }
}

// ---------- Kernel B: gate precompute  xW[dir][b*T+t][4H] = x @ W_x + bias ----------
// Block-tiled: 8 waves produce a 128x64 tile; A/B panels staged in LDS by the TDM,
// double-buffered (K chunks of 20, 300 = 15*20), s_wait_tensorcnt retires the old chunk.
#define KC 20
#define NC 15
__global__ __launch_bounds__(256) void k_xw(const float* __restrict__ x,
                     const float* __restrict__ Wf, const float* __restrict__ bf,
                     const float* __restrict__ Wb, const float* __restrict__ bb,
                     float* __restrict__ xw) {
  int dir = blockIdx.z;
  const float* W    = dir ? Wb : Wf;   // rows 0..299 = W_x of [556][1024] row-major
  const float* bias = dir ? bb : bf;
  int mblk = blockIdx.x;               // 16 row-blocks of 128
  int nblk = blockIdx.y;               // 16 col-blocks of 64
  int lane = threadIdx.x & 31;
  int w    = threadIdx.x >> 5;         // 8 waves
  int n0   = nblk * 64;

  __shared__ float As[2][128 * KC];    // 2 x 10.0 KB
  __shared__ float Bs[2][KC * 64];     // 2 x  5.0 KB

  v8f acc[4];
  for (int j = 0; j < 4; ++j) for (int r = 0; r < 8; ++r) acc[j][r] = 0.0f;

  if (w == 0) {  // wave 0 drives the DMA engine (TDM ignores EXEC, issues per wave)
    tdm_load_2d(lds_off_of(&As[0][0]), x + (size_t)mblk * 128 * 300,
                300, 2048, KC, 128, 300);
    tdm_load_2d(lds_off_of(&Bs[0][0]), W + n0,
                1024, 556, 64, KC, 1024);
  }
  int buf = 0;
  for (int c = 0; c < NC; ++c) {
    if (w == 0) {
      if (c + 1 < NC) {
        tdm_load_2d(lds_off_of(&As[buf ^ 1][0]),
                    x + (size_t)mblk * 128 * 300 + (c + 1) * KC,
                    300, 2048, KC, 128, 300);
        tdm_load_2d(lds_off_of(&Bs[buf ^ 1][0]),
                    W + (size_t)(c + 1) * KC * 1024 + n0,
                    1024, 556, 64, KC, 1024);
        __builtin_amdgcn_s_wait_tensorcnt(2);   // chunk c landed; c+1 still in flight
      } else {
        __builtin_amdgcn_s_wait_tensorcnt(0);
      }
    }
    __syncthreads();                            // chunk c visible to all waves
    const float* Ap = &As[buf][0];
    const float* Bp = &Bs[buf][0];
#pragma unroll
    for (int kk = 0; kk < KC; kk += 4) {
      v2f a = load_a4(Ap, KC, w * 16, kk, lane);
#pragma unroll
      for (int j = 0; j < 4; ++j) {
        v2f b = load_b4(Bp, 64, kk, 16 * j, lane);
        acc[j] = wmma4(a, b, acc[j]);
      }
    }
    __syncthreads();                            // all reads done before buf is rewritten
    buf ^= 1;
  }

  float* outb = xw + (size_t)dir * 2048u * 1024u;
  int half = lane >> 4, cc = lane & 15;
  for (int j = 0; j < 4; ++j) {
    int col = n0 + 16 * j + cc;
    float bv = bias[col];
#pragma unroll
    for (int r = 0; r < 8; ++r) {
      int row = mblk * 128 + w * 16 + half * 8 + r;
      outb[(size_t)row * 1024 + col] = acc[j][r] + bv;
    }
  }
}

// ---------- Kernel C: sequential bi-LSTM recurrence (1 WGP per direction) ----------
__global__ __launch_bounds__(1024) void k_lstm(
    const float* __restrict__ Wf, const float* __restrict__ Wb,
    const float* __restrict__ xw, const int* __restrict__ len,
    float* __restrict__ outbuf) {
  int dir = blockIdx.x;
  const float* Wh  = (dir ? Wb : Wf) + 300 * 1024;            // recurrent rows
  const float* xwd = xw + (size_t)dir * 2048u * 1024u;
  int lane = threadIdx.x & 31;
  int w    = threadIdx.x >> 5;    // 32 waves
  int m0 = (w >> 4) * 16;         // batch tile: rows 0-15 or 16-31
  int u0 = (w & 15) * 16;         // 16-unit slice; wave owns all 4 gates of it
  __shared__ float h_s[32][256];
  __shared__ int   len_s[32];
  for (int i = threadIdx.x; i < 32 * 256; i += 1024) (&h_s[0][0])[i] = 0.0f;
  if (threadIdx.x < 32) len_s[threadIdx.x] = len[threadIdx.x];
  float creg[8];                   // cell state lives in registers (exclusive ownership)
#pragma unroll
  for (int r = 0; r < 8; ++r) creg[r] = 0.0f;
  __syncthreads();
  int half = lane >> 4, cc = lane & 15;
  int col = u0 + cc;
  for (int s = 0; s < 64; ++s) {
    int t = dir ? (63 - s) : s;
    v8f acc[4];
    for (int g = 0; g < 4; ++g) for (int r = 0; r < 8; ++r) acc[g][r] = 0.0f;
    for (int k = 0; k < 256; k += 4) {
      v2f a = load_a4(&h_s[0][0], 256, m0, k, lane);   // h from LDS
#pragma unroll
      for (int g = 0; g < 4; ++g) {                    // i, j, f, o columns
        v2f b = load_b4(Wh, 1024, k, g * 256 + u0, lane);
        acc[g] = wmma4(a, b, acc[g]);
      }
    }
    __syncthreads();   // all waves finished reading h for this step
#pragma unroll
    for (int r = 0; r < 8; ++r) {
      int row = m0 + half * 8 + r;                     // batch index
      size_t zoff = ((size_t)row * 64 + t) * 1024;
      float zi = acc[0][r] + xwd[zoff + col];
      float zj = acc[1][r] + xwd[zoff + 256 + col];
      float zf = acc[2][r] + xwd[zoff + 512 + col];
      float zo = acc[3][r] + xwd[zoff + 768 + col];
      float co = creg[r];
      float cn = sigf(zf + 1.0f) * co + sigf(zi) * tanhf(zj);  // forget_bias = 1.0
      float hn = sigf(zo) * tanhf(cn);
      bool mv = t < len_s[row];
      creg[r] = mv ? cn : co;
      float ho = h_s[row][col];
      h_s[row][col] = mv ? hn : ho;
      outbuf[((size_t)row * 64 + t) * 512 + dir * 256 + col] = mv ? hn : 0.0f;
    }
    __syncthreads();   // h update visible before next step's GEMM
  }
}

// ---------- Kernel D1/D2: einsum 'mbtd,btd->md' / T  (memory-bound, deterministic 2-stage) ----------
__global__ void k_ein1(const float* __restrict__ ent, const float* __restrict__ outbuf,
                       float* __restrict__ part) {
  int m = blockIdx.x, cb = blockIdx.y, d = threadIdx.x;      // 64 x 16 blocks, 512 threads
  const float* e = ent + ((size_t)m * 2048 + (size_t)cb * 128) * 512 + d;
  const float* o = outbuf + (size_t)cb * 128 * 512 + d;
  float sum = 0.0f;
  for (int i = 0; i < 128; ++i) {
    if (((i & 7) == 0) && (i + 16 < 128))
      __builtin_prefetch(e + (size_t)(i + 16) * 512, 0, 1);  // global_prefetch_b8 ahead of stream
    sum += e[(size_t)i * 512] * o[(size_t)i * 512];
  }
  part[((size_t)m * 16 + cb) * 512 + d] = sum;
}
__global__ void k_ein2(const float* __restrict__ part, float* __restrict__ outf) {
  int m = blockIdx.x, d = threadIdx.x;
  float s = 0.0f;
  for (int cb = 0; cb < 16; ++cb) s += part[((size_t)m * 16 + cb) * 512 + d];
  outf[(size_t)m * 512 + d] = s * (1.0f / 64.0f);
}

// ---------- Kernel E: relu(outf@W1+b1) @ W2 + b2 -> sigmoid, prediction + loss ----------
__global__ __launch_bounds__(512) void k_head(
    const float* __restrict__ outf, const float* __restrict__ W1, const float* __restrict__ b1,
    const float* __restrict__ W2, const float* __restrict__ b2,
    const int* __restrict__ labels, float* __restrict__ dout) {
  __shared__ float l1_s[64][256];    // 64 KB
  int lane = threadIdx.x & 31, w = threadIdx.x >> 5;   // 16 waves
  int m0 = (w >> 2) * 16;
  int n0 = (w & 3) * 64;
  v8f acc[4];
  for (int j = 0; j < 4; ++j) for (int r = 0; r < 8; ++r) acc[j][r] = 0.0f;
  for (int k = 0; k < 512; k += 4) {
    v2f a = load_a4(outf, 512, m0, k, lane);
#pragma unroll
    for (int j = 0; j < 4; ++j) {
      v2f b = load_b4(W1, 256, k, n0 + 16 * j, lane);
      acc[j] = wmma4(a, b, acc[j]);
    }
  }
  int half = lane >> 4, cc = lane & 15;
  for (int j = 0; j < 4; ++j) {
    int colc = n0 + 16 * j + cc;
    float bv = b1[colc];
#pragma unroll
    for (int r = 0; r < 8; ++r) {
      float v = acc[j][r] + bv;
      l1_s[m0 + half * 8 + r][colc] = v > 0.0f ? v : 0.0f;   // relu
    }
  }
  __syncthreads();
  float contrib = 0.0f;
  if (threadIdx.x < 320) {
    int m = threadIdx.x / 5, c = threadIdx.x % 5;
    float xv = b2[c];
    for (int k = 0; k < 256; ++k) xv += l1_s[m][k] * W2[k * 5 + c];
    float p = 1.0f / (1.0f + expf(-xv));
    dout[m * 5 + c] = p;
    contrib = -(float)labels[m * 5 + c] * logf(p);
  }
  __syncthreads();                    // l1_s no longer needed; reuse for reduction
  float* red = &l1_s[0][0];
  if (threadIdx.x < 320) red[threadIdx.x] = contrib;
  __syncthreads();
  if (threadIdx.x == 0) {
    float s = 0.0f;
    for (int i = 0; i < 320; ++i) s += red[i];   // fixed order -> deterministic
    dout[320] = s * (1.0f / 5.0f);               // mean over C=5 of column sums
  }
}

extern "C" void kernel_launch(void* const* d_in, const int* in_sizes, int n_in,
                              void* d_out, int out_size, void* d_ws, size_t ws_size,
                              hipStream_t stream) {
  (void)in_sizes; (void)n_in; (void)out_size; (void)ws_size;
  const float* x   = (const float*)d_in[0];
  const float* ent = (const float*)d_in[1];
  const int*   lab = (const int*)  d_in[2];
  const float* Wf  = (const float*)d_in[3];
  const float* bf  = (const float*)d_in[4];
  const float* Wb  = (const float*)d_in[5];
  const float* bb  = (const float*)d_in[6];
  const float* W1  = (const float*)d_in[7];
  const float* b1  = (const float*)d_in[8];
  const float* W2  = (const float*)d_in[9];
  const float* b2  = (const float*)d_in[10];
  float* ws     = (float*)d_ws;
  float* xw     = ws + WS_XW;
  float* outbuf = ws + WS_OUTBUF;
  float* part   = ws + WS_PART;
  float* outf   = ws + WS_OUTF;
  int*   len    = (int*)(ws + WS_LEN);
  float* dout   = (float*)d_out;

  k_len <<<dim3(32),         dim3(64),   0, stream>>>(x, len);
  k_xw  <<<dim3(16, 16, 2),  dim3(256),  0, stream>>>(x, Wf, bf, Wb, bb, xw);
  k_lstm<<<dim3(2),          dim3(1024), 0, stream>>>(Wf, Wb, xw, len, outbuf);
  k_ein1<<<dim3(64, 16),     dim3(512),  0, stream>>>(ent, outbuf, part);
  k_ein2<<<dim3(64),         dim3(512),  0, stream>>>(part, outf);
  k_head<<<dim3(1),          dim3(512),  0, stream>>>(outf, W1, b1, W2, b2, lab, dout);
}